// SVPF_69655779607071
// MI455X (gfx1250) — compile-verified
//
#include <hip/hip_runtime.h>
#include <hip/hip_bf16.h>

typedef __attribute__((ext_vector_type(2))) float v2f;
typedef __attribute__((ext_vector_type(8))) float v8f;

#define NPART   500
#define NPAD    512
#define TLEN    512
#define NSTEIN  10
#define MED_TARGET 62375             /* smallest v with count(pairs<=v) >= this */
#define BW_DIV  6.2166061010848646f  /* log(501) */
#define GLN_CONST (-0.96861964f)     /* gammaln(3)-gammaln(2.5)-0.5*log(5*pi)  */
#define LOG2E   1.4426950408889634f

__device__ __forceinline__ float clampf(float v, float lo, float hi) {
    return fminf(fmaxf(v, lo), hi);
}

__device__ __forceinline__ float waveSum(float v) {
#pragma unroll
    for (int o = 16; o; o >>= 1) v += __shfl_xor(v, o, 32);
    return v;
}
__device__ __forceinline__ float waveMax(float v) {
#pragma unroll
    for (int o = 16; o; o >>= 1) v = fmaxf(v, __shfl_xor(v, o, 32));
    return v;
}
__device__ __forceinline__ int waveSumI(int v) {
#pragma unroll
    for (int o = 16; o; o >>= 1) v += __shfl_xor(v, o, 32);
    return v;
}

__device__ float blockSum(float v, float* red) {
    int lane = threadIdx.x & 31, wv = threadIdx.x >> 5;
    v = waveSum(v);
    if (!lane) red[wv] = v;
    __syncthreads();
    float tot = 0.f;
#pragma unroll
    for (int i = 0; i < 16; ++i) tot += red[i];
    __syncthreads();
    return tot;
}
__device__ float blockMax(float v, float* red) {
    int lane = threadIdx.x & 31, wv = threadIdx.x >> 5;
    v = waveMax(v);
    if (!lane) red[wv] = v;
    __syncthreads();
    float tot = -__builtin_inff();
#pragma unroll
    for (int i = 0; i < 16; ++i) tot = fmaxf(tot, red[i]);
    __syncthreads();
    return tot;
}

// intra-wave bitonic compare-exchange (stride j < 32) via lane shuffle
__device__ __forceinline__ float cmpx(float v, int j, bool up) {
    float pv = __shfl_xor(v, j, 32);
    bool lower = ((threadIdx.x & j) == 0);
    return (lower == up) ? fminf(v, pv) : fmaxf(v, pv);
}

__global__ __launch_bounds__(512) void svpf_kernel(
    const float* __restrict__ p_log_rho,
    const float* __restrict__ p_log_sigma,
    const float* __restrict__ p_mu,
    const float* __restrict__ obs,
    const float* __restrict__ init_noise,
    const float* __restrict__ noise,
    float* __restrict__ out)
{
    __shared__ float xb0[NPAD], xb1[NPAD];   // double-buffered particles
    __shared__ float hp[NPAD];               // h_prev for this scan step
    __shared__ float gs[NPAD];               // SVGD gradients
    __shared__ float srt[NPAD];              // sort scratch for median
    __shared__ float red[16];                // block-reduce scratch
    __shared__ float bcast;                  // bandwidth broadcast

    const int tid  = threadIdx.x;
    const int lane = tid & 31;
    const int wv   = tid >> 5;

    const float log_rho   = p_log_rho[0];
    const float log_sigma = p_log_sigma[0];
    const float mu        = p_mu[0];
    const float rho       = 0.999f / (1.f + __expf(-log_rho));
    const float sigma_z   = __expf(log_sigma);
    const float sig2      = sigma_z * sigma_z;
    const float stat_sd   = sqrtf(sig2 / (1.f - rho * rho + 1e-6f));
    const float invN      = 1.f / (float)NPART;

    // lane-constant A-matrix row-selector weights (rows: 0=g, 1=ones, 2=x)
    const int   mrow = lane & 15;
    const int   koff = (lane >> 4) << 1;          // i-chunk sub-offset: 0 or 2
    const float wg = (mrow == 0) ? 1.f : 0.f;
    const float wc = (mrow == 1) ? 1.f : 0.f;
    const float wx = (mrow == 2) ? 1.f : 0.f;

    xb0[tid] = (tid < NPART) ? (mu + stat_sd * init_noise[tid]) : 0.f;
    __syncthreads();

    float ll = 0.f;

    for (int t = 0; t < TLEN; ++t) {
        const float y  = obs[t];
        const float y2 = y * y;

        // ---------------- predict + log-weights ----------------
        float logw = -__builtin_inff();
        {
            float hprev = xb0[tid];
            hp[tid] = hprev;
            if (tid < NPART) {
                float eps   = noise[t * NPART + tid];
                float hpred = clampf(mu + rho * (hprev - mu) + sigma_z * eps,
                                     -15.f, 5.f);
                float s = y2 / (__expf(hpred) + 1e-8f);
                logw = GLN_CONST - 0.5f * hpred - 3.0f * log1pf(s * 0.2f);
                xb0[tid] = hpred;
                if (t + 1 < TLEN)
                    __builtin_prefetch(&noise[(t + 1) * NPART + tid], 0, 3);
            }
        }
        __syncthreads();

        float m  = blockMax(logw, red);
        float w  = (tid < NPART) ? __expf(logw - m) : 0.f;
        float sw = blockSum(w, red);
        if (tid == 0) ll += m + __logf(sw * invN + 1e-10f);

        // ---------------- SVGD inner loop ----------------
        for (int s = 0; s < NSTEIN; ++s) {
            float* xs = (s & 1) ? xb1 : xb0;
            float* xd = (s & 1) ? xb0 : xb1;

            // gradient of log posterior; stage sort value in a register
            float sv_elem;
            {
                float xc = xs[tid];
                if (tid < NPART) {
                    float h_c  = clampf(xc, -15.f, 5.f);
                    float hp_c = clampf(hp[tid], -15.f, 5.f);
                    float mu_prior   = mu + rho * (hp_c - mu);
                    float grad_prior = -(h_c - mu_prior) / (sig2 + 1e-6f);
                    float sv = clampf(y2 / (__expf(h_c) + 1e-8f), 0.f, 1e6f);
                    float grad_lik = 0.5f * (6.f * sv / (5.f + sv + 1e-8f) - 1.f);
                    gs[tid] = clampf(grad_prior + grad_lik, -10.f, 10.f);
                    sv_elem = xc;
                } else {
                    gs[tid] = 0.f;
                    sv_elem = __builtin_inff();   // pads sort to the end
                }
            }

            // ---- bitonic sort of 512, wave32-aware ----
            // k = 2..32: fully intra-wave (shuffles, no barriers)
            {
                float v = sv_elem;
#pragma unroll
                for (int k = 2; k <= 32; k <<= 1) {
                    bool up = ((tid & k) == 0);
#pragma unroll
                    for (int j = k >> 1; j > 0; j >>= 1)
                        v = cmpx(v, j, up);
                }
                srt[tid] = v;
            }
            __syncthreads();
            // k = 64..512: j>=32 via LDS, j<=16 back in registers
            for (int k = 64; k <= NPAD; k <<= 1) {
                bool up = ((tid & k) == 0);
                for (int j = k >> 1; j >= 32; j >>= 1) {
                    int ixj = tid ^ j;
                    if (ixj > tid) {
                        float a = srt[tid], b = srt[ixj];
                        if ((a > b) == up) { srt[tid] = b; srt[ixj] = a; }
                    }
                    __syncthreads();
                }
                {
                    float v = srt[tid];
#pragma unroll
                    for (int j = 16; j > 0; j >>= 1)
                        v = cmpx(v, j, up);
                    srt[tid] = v;
                }
                __syncthreads();
            }

            // exact median via bisection on float bit patterns (wave 0)
            if (wv == 0) {
                float dmax = srt[NPART - 1] - srt[0];
                unsigned lo = 0u, hi = __float_as_uint(fmaxf(dmax, 0.f));
                while (lo < hi) {
                    unsigned mid = lo + ((hi - lo) >> 1);
                    float tval = __uint_as_float(mid);
                    int cnt = 0;
                    for (int i = lane; i < NPART; i += 32) {
                        float xi = srt[i];
                        int a = i + 1, b = NPART;
                        while (a < b) {              // first idx with d > tval
                            int mm = (a + b) >> 1;
                            if (srt[mm] - xi <= tval) a = mm + 1; else b = mm;
                        }
                        cnt += a - 1 - i;
                    }
                    cnt = waveSumI(cnt);
                    if (cnt >= MED_TARGET) hi = mid; else lo = mid + 1;
                }
                if (lane == 0)
                    bcast = clampf(__uint_as_float(lo) / BW_DIV, 0.01f, 10.f);
            }
            __syncthreads();

            const float bw      = bcast;
            const float inv_bw2 = 1.f / (bw * bw);
            const float c2l     = -0.5f * inv_bw2 * LOG2E;  // exp2 coefficient

            // ---- all-pairs interaction via V_WMMA_F32_16X16X4_F32 ----
            // Per wave: 16 j-columns.  B = 4x16 tile of K (2 exp2/lane).
            // A rows: 0 = g[i-chunk], 1 = ones, 2 = x[i-chunk]  =>
            // D row0 = K^T g, row1 = sum K, row2 = sum K*x_i, each landing in
            // lane (j - jbase) of acc[0..2].
#pragma unroll
            for (int p = 0; p < 2; ++p) {
                const int jbase = ((p << 4) + wv) << 4;
                const int jcol  = jbase + mrow;
                const float xj  = xs[jcol];
                v8f acc = {0.f, 0.f, 0.f, 0.f, 0.f, 0.f, 0.f, 0.f};
                for (int c = 0; c < NPART / 4; ++c) {
                    const int ib = (c << 2) + koff;
                    float xi0 = xs[ib],     xi1 = xs[ib + 1];
                    float gi0 = gs[ib],     gi1 = gs[ib + 1];
                    float d0 = xj - xi0,    d1 = xj - xi1;
                    float b0 = __builtin_amdgcn_exp2f(c2l * d0 * d0);
                    float b1 = __builtin_amdgcn_exp2f(c2l * d1 * d1);
                    float a0 = fmaf(wg, gi0, fmaf(wx, xi0, wc));
                    float a1 = fmaf(wg, gi1, fmaf(wx, xi1, wc));
                    v2f A = {a0, a1};
                    v2f B = {b0, b1};
                    acc = __builtin_amdgcn_wmma_f32_16x16x4_f32(
                        false, A, false, B, (short)0, acc, false, false);
                }
                if (lane < 16) {
                    float att = acc[0];                       // (K^T g)[j]
                    float s1  = acc[1];                       // sum_i K
                    float s2  = acc[2];                       // sum_i K*x_i
                    float rep = -(xj * s1 - s2) * inv_bw2;    // sum_i gradK
                    float upd = clampf(0.1f * (att + rep) * invN, -1.f, 1.f);
                    xd[jcol] = (jcol < NPART) ? clampf(xj + upd, -15.f, 5.f)
                                              : 0.f;
                }
            }
            __syncthreads();
        }

        // volatility mean (10 SVGD steps -> final state back in xb0)
        float vol = (tid < NPART) ? __expf(xb0[tid] * 0.5f) : 0.f;
        float vs  = blockSum(vol, red);
        if (tid == 0) out[1 + t] = vs * invN;
        __syncthreads();
    }
    if (tid == 0) out[0] = ll;
}

extern "C" void kernel_launch(void* const* d_in, const int* in_sizes, int n_in,
                              void* d_out, int out_size, void* d_ws, size_t ws_size,
                              hipStream_t stream) {
    (void)in_sizes; (void)n_in; (void)out_size; (void)d_ws; (void)ws_size;
    const float* log_rho    = (const float*)d_in[0];
    const float* log_sigma  = (const float*)d_in[1];
    const float* mu         = (const float*)d_in[2];
    const float* obs        = (const float*)d_in[3];
    const float* init_noise = (const float*)d_in[4];
    const float* noise      = (const float*)d_in[5];
    float* out = (float*)d_out;
    svpf_kernel<<<1, NPAD, 0, stream>>>(log_rho, log_sigma, mu,
                                        obs, init_noise, noise, out);
}